// SparseLayerWithExternalBkg_56659208568922
// MI455X (gfx1250) — compile-verified
//
#include <hip/hip_runtime.h>
#include <stdint.h>

// Problem constants (match reference).
#define T_STEPS   128
#define R_ROWS    200000
#define RPB       64          // rows per block: 200000 / 64 = 3125 exactly
#define OUT_PITCH 129         // 128 t's + 1 pad -> conflict-free LDS transpose
#define BKG_MAX_C 100         // bkg source neurons (s_bkg sized for this)
#define CHUNK     512         // nonzeros staged into LDS per round

typedef uint32_t u32x4 __attribute__((ext_vector_type(4)));
typedef int      i32x8 __attribute__((ext_vector_type(8)));
typedef int      i32x4 __attribute__((ext_vector_type(4)));

__device__ __forceinline__ int lower_bound_i32(const int* __restrict__ a, int n, int key) {
    int lo = 0, hi = n;
    while (lo < hi) {
        int mid = (lo + hi) >> 1;
        if (a[mid] < key) lo = mid + 1; else hi = mid;
    }
    return lo;
}

// Per-lane async copy of `cnt` 4-byte elements src[base..base+cnt) -> LDS
// (ASYNCcnt machinery; also the fallback when the TDM builtin is absent).
__device__ __forceinline__ void async_stage_b32(uint32_t ldsByteOff, const void* src,
                                                int base, int cnt, int tid) {
    for (int u = tid; u < cnt; u += 128) {
        uint32_t la = ldsByteOff + (uint32_t)u * 4u;
        uint32_t go = (uint32_t)(base + u) * 4u;
        asm volatile("global_load_async_to_lds_b32 %0, %1, %2"
                     :: "v"(la), "v"(go), "s"(src) : "memory");
    }
}

#if __has_builtin(__builtin_amdgcn_tensor_load_to_lds)
#define HAVE_TDM 1
// Issue one TDM descriptor: contiguous 1-row tile of nElems 4-byte elements,
// global (src + elemBase*4) -> LDS byte offset ldsByteOff. Caller restricts
// execution to a single wave; completion via s_wait_tensorcnt.
__device__ __forceinline__ void tdm_load_1d(uint32_t ldsByteOff, const void* src,
                                            int elemBase, int nElems) {
    const uint64_t ga = (uint64_t)(uintptr_t)src + (uint64_t)(uint32_t)elemBase * 4ull;
    const uint32_t ne = (uint32_t)nElems;                 // <= 512 (fits tile_dim0)
    u32x4 g0;
    g0[0] = 1u;                                           // count=1 (valid user D#)
    g0[1] = ldsByteOff;                                   // lds_addr
    g0[2] = (uint32_t)ga;                                 // global_addr[31:0]
    g0[3] = (uint32_t)((ga >> 32) & 0x1FFFFFFu) | (2u << 30); // addr[56:32] | type=2
    i32x8 g1;
    g1[0] = (int)(2u << 16);                              // wg_mask=0, data_size=4B
    g1[1] = (int)((ne & 0xFFFFu) << 16);                  // tensor_dim0[15:0]
    g1[2] = (int)((ne >> 16) | (1u << 16));               // tensor_dim0[31:16] | tensor_dim1=1
    g1[3] = (int)((ne & 0xFFFFu) << 16);                  // tile_dim0 = ne
    g1[4] = 1;                                            // tile_dim1 = 1
    g1[5] = (int)ne;                                      // tensor_dim0_stride[31:0]
    g1[6] = (int)(ne << 16);                              // tensor_dim1_stride[15:0]
    g1[7] = 0;
    i32x4 g2 = {0, 0, 0, 0};
    i32x4 g3 = {0, 0, 0, 0};
#if defined(__clang_major__) && (__clang_major__ >= 23)
    i32x8 g4 = {0, 0, 0, 0, 0, 0, 0, 0};
    __builtin_amdgcn_tensor_load_to_lds(g0, g1, g2, g3, g4, 0);
#else
    __builtin_amdgcn_tensor_load_to_lds(g0, g1, g2, g3, 0);
#endif
}
__device__ __forceinline__ void tdm_wait0() {
#if __has_builtin(__builtin_amdgcn_s_wait_tensorcnt)
    __builtin_amdgcn_s_wait_tensorcnt((short)0);
#else
    asm volatile("s_wait_tensorcnt 0" ::: "memory");
#endif
}
#endif

// Stage one chunk of the nonzero stream (rows/cols/weights) into LDS.
// TDM path: 3 descriptors issued by wave 0; fallback: per-lane async b32.
__device__ __forceinline__ void stage_chunk(uint32_t off_rows, uint32_t off_cols,
                                            uint32_t off_w,
                                            const int* rows, const int* cols,
                                            const float* w,
                                            int base, int cnt, int tid) {
#ifdef HAVE_TDM
    if (tid < 32) {                        // wave 0 drives the DMA engine
        tdm_load_1d(off_rows, rows, base, cnt);
        tdm_load_1d(off_cols, cols, base, cnt);
        tdm_load_1d(off_w,    w,    base, cnt);
    }
    tdm_wait0();                           // waves 1-3 pass trivially
#else
    async_stage_b32(off_rows, rows, base, cnt, tid);
    async_stage_b32(off_cols, cols, base, cnt, tid);
    async_stage_b32(off_w,    w,    base, cnt, tid);
    asm volatile("s_wait_asynccnt 0" ::: "memory");
#endif
}

// Pack binary spike matrix (T=128, C) into transposed bitmasks:
// packed[c*4 + w] bit l  ==  spikes[(w*32+l)*C + c] != 0.
__global__ void pack_spikes_kernel(const float* __restrict__ spikes, int C,
                                   uint32_t* __restrict__ packed) {
    __shared__ float tile[T_STEPS * 33];
    const int tid = threadIdx.x;           // 128 threads
    const int c0  = blockIdx.x * 32;

    for (int idx = tid; idx < T_STEPS * 32; idx += 128) {
        int t = idx >> 5, cl = idx & 31;
        int c = c0 + cl;
        tile[t * 33 + cl] = (c < C) ? __builtin_nontemporal_load(&spikes[t * C + c]) : 0.0f;
    }
    __syncthreads();

    const int lane = tid & 31, wave = tid >> 5;
    for (int cl = 0; cl < 32; ++cl) {
        bool bit = tile[tid * 33 + cl] != 0.0f;
        uint32_t m = __builtin_amdgcn_ballot_w32(bit);
        if (lane == 0) {
            int c = c0 + cl;
            if (c < C) packed[c * 4 + wave] = m;
        }
    }
}

// One block = 64 consecutive output rows, 128 threads = the 128 timesteps.
__global__ void spmm_kernel(const uint32_t* __restrict__ lgnPacked,
                            const uint32_t* __restrict__ bkgPacked, int C_B,
                            const int* __restrict__ lrows,
                            const int* __restrict__ lcols,
                            const float* __restrict__ lw, int nnzL,
                            const int* __restrict__ brows,
                            const int* __restrict__ bcols,
                            const float* __restrict__ bw, int nnzB,
                            float* __restrict__ out) {
    __shared__ uint32_t s_bkg[BKG_MAX_C * 4];       // 1.6 KB bkg bitmask
    __shared__ int      s_rows[CHUNK];              // staged nonzero stream
    __shared__ int      s_cols[CHUNK];
    __shared__ float    s_w[CHUNK];
    __shared__ float    s_out[RPB * OUT_PITCH];     // 33 KB output tile

    const int tid  = threadIdx.x;          // tid == timestep t
    const int lane = tid & 31;
    const int wave = tid >> 5;
    const int r0   = blockIdx.x * RPB;

    // BKG bitmask -> LDS via per-lane async copy (ASYNCcnt family).
    if (tid < C_B) {
        uint32_t la = (uint32_t)(uintptr_t)(&s_bkg[tid * 4]);
        uint32_t go = (uint32_t)(tid * 16);
        asm volatile("global_load_async_to_lds_b128 %0, %1, %2"
                     :: "v"(la), "v"(go), "s"(bkgPacked) : "memory");
    }

    // Overlap with the DMA: binary-search this block's nonzero ranges.
    const int jl0 = lower_bound_i32(lrows, nnzL, r0);
    const int jl1 = lower_bound_i32(lrows, nnzL, r0 + RPB);
    const int jb0 = lower_bound_i32(brows, nnzB, r0);
    const int jb1 = lower_bound_i32(brows, nnzB, r0 + RPB);

    // Zero the output tile (each thread owns its own t column -> race free).
    for (int rl = 0; rl < RPB; ++rl) s_out[rl * OUT_PITCH + tid] = 0.0f;

    asm volatile("s_wait_asynccnt 0" ::: "memory");
    __syncthreads();

    const uint32_t off_rows = (uint32_t)(uintptr_t)(&s_rows[0]);
    const uint32_t off_cols = (uint32_t)(uintptr_t)(&s_cols[0]);
    const uint32_t off_w    = (uint32_t)(uintptr_t)(&s_w[0]);

    // ---- LGN source: TDM-staged chunks, segmented row accumulation ----
    {
        int   rcur = -1;
        float acc  = 0.0f;
        for (int base = jl0; base < jl1; base += CHUNK) {
            const int cnt = min(CHUNK, jl1 - base);
            __syncthreads();                        // buffer reuse guard
            stage_chunk(off_rows, off_cols, off_w, lrows, lcols, lw, base, cnt, tid);
            __builtin_prefetch(lrows + base + CHUNK, 0, 0);   // global_prefetch_b8
            __syncthreads();
            for (int k = 0; k < cnt; ++k) {
                const int r = s_rows[k];
                if (r != rcur) {
                    if (rcur >= 0) s_out[(rcur - r0) * OUT_PITCH + tid] += acc;
                    rcur = r; acc = 0.0f;
                }
                const uint32_t m = lgnPacked[(s_cols[k] << 2) + wave]; // uniform 4B L2 read
                acc += ((m >> lane) & 1u) ? s_w[k] : 0.0f;             // exact: spikes 0/1
            }
        }
        if (rcur >= 0) s_out[(rcur - r0) * OUT_PITCH + tid] += acc;
    }

    // ---- BKG source: same scheme, bitmask lives in LDS ----
    {
        int   rcur = -1;
        float acc  = 0.0f;
        for (int base = jb0; base < jb1; base += CHUNK) {
            const int cnt = min(CHUNK, jb1 - base);
            __syncthreads();
            stage_chunk(off_rows, off_cols, off_w, brows, bcols, bw, base, cnt, tid);
            __syncthreads();
            for (int k = 0; k < cnt; ++k) {
                const int r = s_rows[k];
                if (r != rcur) {
                    if (rcur >= 0) s_out[(rcur - r0) * OUT_PITCH + tid] += acc;
                    rcur = r; acc = 0.0f;
                }
                const uint32_t m = s_bkg[(s_cols[k] << 2) + wave];     // LDS broadcast
                acc += ((m >> lane) & 1u) ? s_w[k] : 0.0f;
            }
        }
        if (rcur >= 0) s_out[(rcur - r0) * OUT_PITCH + tid] += acc;
    }
    __syncthreads();

    // Transpose write-out: each wave stores contiguous 128B runs of
    // out[t*R + r0 .. r0+63]; write-once stream -> non-temporal.
    for (int idx = tid; idx < RPB * T_STEPS; idx += 128) {
        int t = idx >> 6;                  // idx = t*64 + i
        int i = idx & 63;
        __builtin_nontemporal_store(s_out[i * OUT_PITCH + t],
                                    &out[(size_t)t * R_ROWS + (size_t)(r0 + i)]);
    }
}

extern "C" void kernel_launch(void* const* d_in, const int* in_sizes, int n_in,
                              void* d_out, int out_size, void* d_ws, size_t ws_size,
                              hipStream_t stream) {
    const float* lgn_spikes = (const float*)d_in[0];   // (1,128,17400)
    const float* bkg_spikes = (const float*)d_in[1];   // (1,128,100)
    const int*   lrows      = (const int*)  d_in[2];
    const int*   lcols      = (const int*)  d_in[3];
    const float* lw         = (const float*)d_in[4];
    const int*   brows      = (const int*)  d_in[5];
    const int*   bcols      = (const int*)  d_in[6];
    const float* bw         = (const float*)d_in[7];

    const int nnzL = in_sizes[2];
    const int nnzB = in_sizes[5];
    const int C_L  = in_sizes[0] / T_STEPS;            // 17400
    const int C_B  = in_sizes[1] / T_STEPS;            // 100

    float* out = (float*)d_out;

    // Workspace: [ lgnPacked : C_L*4 u32 (278 KB) | bkgPacked : C_B*4 u32 ]
    uint32_t* lgnPacked = (uint32_t*)d_ws;
    uint32_t* bkgPacked = lgnPacked + (size_t)C_L * 4;

    pack_spikes_kernel<<<(C_L + 31) / 32, 128, 0, stream>>>(lgn_spikes, C_L, lgnPacked);
    pack_spikes_kernel<<<(C_B + 31) / 32, 128, 0, stream>>>(bkg_spikes, C_B, bkgPacked);

    spmm_kernel<<<R_ROWS / RPB, 128, 0, stream>>>(
        lgnPacked, bkgPacked, C_B,
        lrows, lcols, lw, nnzL,
        brows, bcols, bw, nnzB,
        out);
}